// GraphSAGECF_55860344651847
// MI455X (gfx1250) — compile-verified
//
#include <hip/hip_runtime.h>
#include <hip/hip_bf16.h>

#define N_USERS 200000
#define N_ITEMS 100000
#define DIM 64
#define K2 128           // 2*DIM (concat width)
#define N_LAYERS 2
#define N_EDGES 3200000

typedef __attribute__((ext_vector_type(2))) float v2f;
typedef __attribute__((ext_vector_type(8))) float v8f;

// ---------------------------------------------------------------------------
// Zero a float buffer (float4 granularity; counts here are multiples of 4).
// ---------------------------------------------------------------------------
__global__ void zero_kernel(float4* __restrict__ p, long n4) {
    long t = (long)blockIdx.x * blockDim.x + threadIdx.x;
    long stride = (long)gridDim.x * blockDim.x;
    float4 z = make_float4(0.f, 0.f, 0.f, 0.f);
    for (long i = t; i < n4; i += stride) p[i] = z;
}

// ---------------------------------------------------------------------------
// SpMM scatter: out[rows[e]] += vals[e] * H[cols[e]]   (rows of DIM=64 f32)
// 16 lanes cooperate per edge: each lane handles 4 consecutive floats.
// Gather is a coalesced 256B row read; scatter-add hits L2-resident
// accumulators (both fit in the 192MB L2).
// ---------------------------------------------------------------------------
__global__ void spmm_scatter(const float* __restrict__ vals,
                             const int*   __restrict__ rows,
                             const int*   __restrict__ cols,
                             const float* __restrict__ H,
                             float*       __restrict__ Out,
                             int nedges) {
    long t = (long)blockIdx.x * blockDim.x + threadIdx.x;
    int  lane = (int)(t & 15);
    long g = t >> 4;
    long ngroups = ((long)gridDim.x * blockDim.x) >> 4;
    for (long e = g; e < nedges; e += ngroups) {
        float w = vals[e];
        long src = cols[e];
        long dst = rows[e];
        const float4 h4 = *(const float4*)(H + src * DIM + lane * 4);
        float* o = Out + dst * DIM + lane * 4;
        atomicAdd(o + 0, w * h4.x);
        atomicAdd(o + 1, w * h4.y);
        atomicAdd(o + 2, w * h4.z);
        atomicAdd(o + 3, w * h4.w);
    }
}

// ---------------------------------------------------------------------------
// Dense layer + ReLU + row L2-normalize, built on V_WMMA_F32_16X16X4_F32.
//
//   Hout[m, n] = relu( sum_k concat(Hin, Neigh)[m, k] * W[n, k] ) , then
//   row-normalized by 1/max(||row||2, 1e-12).
//
// Block = 128 threads = 4 waves; block handles 16 rows; wave w owns the
// 16-column tile n in [16w, 16w+16).  K=128 -> 32 WMMA steps of K=4.
//
// WMMA f32 16x16x4 operand layouts (ISA 7.12.2):
//   A (16x4, MxK):  lanes 0-15 hold row M=lane, VGPR0=K0,VGPR1=K1;
//                   lanes 16-31 hold row M=lane-16, VGPR0=K2,VGPR1=K3.
//   B (4x16, KxN):  lanes 0-15 hold col N=lane  (K0 in VGPR0, K1 in VGPR1);
//                   lanes 16-31 hold col N=lane-16 (K2 in VGPR0, K3 in VGPR1).
//   C/D (16x16):    VGPR v: lanes 0-15 -> (M=v, N=lane),
//                            lanes 16-31 -> (M=v+8, N=lane-16).
//
// B[k][n] = W[n][k] (W is [64,128] row-major), so each lane's B fragment for
// step s (k0=4s) is the float2 at W + n*128 + k0 + (lane>=16 ? 2 : 0).
// All 32 B fragments are loaded once per wave and live in registers (64 VGPRs).
// ---------------------------------------------------------------------------
__global__ __launch_bounds__(128)
void dense_norm(const float* __restrict__ Hin,    // [nrows, 64]
                const float* __restrict__ Neigh,  // [nrows, 64]
                const float* __restrict__ W,      // [64, 128] row-major
                float*       __restrict__ Hout,   // [nrows, 64]
                int nrows) {
    __shared__ float smem[16][68];   // padded to dodge bank conflicts
    __shared__ float invn[16];

    const int tid  = threadIdx.x;
    const int lane = tid & 31;
    const int wave = tid >> 5;          // 0..3, owns N-tile [16*wave, 16*wave+16)
    const int n_off = wave * 16;
    const int hi_half = (lane >= 16) ? 1 : 0;
    const int l15 = lane & 15;

    // --- Load all B fragments of W for this wave's N-tile (once) ---
    v2f bfrag[32];
    {
        const int n = n_off + l15;
        const float* wb = W + (long)n * K2 + 2 * hi_half;
#pragma unroll
        for (int s = 0; s < 32; ++s) {
            bfrag[s] = *(const v2f*)(wb + 4 * s);
        }
    }

    const long row0 = (long)blockIdx.x * 16;   // nrows is a multiple of 16

    // --- K loop: 32 x V_WMMA_F32_16X16X4_F32, fp32 accumulate ---
    v8f c = {};
    {
        const long arow = row0 + l15;
        const float* baseH = Hin   + arow * DIM + 2 * hi_half;
        const float* baseN = Neigh + arow * DIM + 2 * hi_half;
#pragma unroll
        for (int s = 0; s < 32; ++s) {
            const int k0 = 4 * s;
            const float* src = (k0 < DIM) ? (baseH + k0) : (baseN + (k0 - DIM));
            v2f a = *(const v2f*)src;
            c = __builtin_amdgcn_wmma_f32_16x16x4_f32(
                    /*neg_a=*/false, a, /*neg_b=*/false, bfrag[s],
                    /*c_mod=*/(short)0, c, /*reuse_a=*/false, /*reuse_b=*/false);
        }
    }

    // --- ReLU + stage 16x16 tile into LDS ---
#pragma unroll
    for (int v = 0; v < 8; ++v) {
        float x = c[v];
        x = fmaxf(x, 0.0f);
        smem[v + 8 * hi_half][n_off + l15] = x;
    }
    __syncthreads();

    // --- Row L2 norms (threads 0-15, one row each; conflict-free banks) ---
    if (tid < 16) {
        float s = 0.f;
#pragma unroll
        for (int cc = 0; cc < DIM; ++cc) {
            float x = smem[tid][cc];
            s += x * x;
        }
        invn[tid] = 1.0f / fmaxf(sqrtf(s), 1e-12f);
    }
    __syncthreads();

    // --- Scaled, coalesced store: 16*64 elems / 128 threads = 8 each ---
#pragma unroll
    for (int i = 0; i < 8; ++i) {
        int idx = tid + 128 * i;
        int r = idx >> 6;
        int col = idx & 63;
        Hout[(row0 + r) * DIM + col] = smem[r][col] * invn[r];
    }
}

// ---------------------------------------------------------------------------
// Orchestration: 2 layers of { zero, spmm_u, spmm_i, dense_u, dense_i }.
// Layer 0 writes ping-pong buffers in d_ws; layer 1 writes d_out directly.
// ---------------------------------------------------------------------------
extern "C" void kernel_launch(void* const* d_in, const int* in_sizes, int n_in,
                              void* d_out, int out_size, void* d_ws, size_t ws_size,
                              hipStream_t stream) {
    const float* user_emb = (const float*)d_in[0];
    const float* item_emb = (const float*)d_in[1];
    const float* Wu       = (const float*)d_in[2];   // [2, 64, 128]
    const float* Wi       = (const float*)d_in[3];
    const int*   u_idx    = (const int*)  d_in[4];
    const int*   i_idx    = (const int*)  d_in[5];
    const float* w_u2i    = (const float*)d_in[6];
    const float* w_i2u    = (const float*)d_in[7];

    float* out_u = (float*)d_out;
    float* out_i = out_u + (long)N_USERS * DIM;

    const long n_u = (long)N_USERS * DIM;   // 12.8M floats
    const long n_i = (long)N_ITEMS * DIM;   //  6.4M floats

    char* ws = (char*)d_ws;
    float* neigh_u = (float*)ws;              ws += n_u * sizeof(float);
    float* neigh_i = (float*)ws;              ws += n_i * sizeof(float);
    float* h_u1    = (float*)ws;              ws += n_u * sizeof(float);
    float* h_i1    = (float*)ws;              /* ws += n_i * sizeof(float); */

    const float* hu = user_emb;
    const float* hi = item_emb;

    for (int l = 0; l < N_LAYERS; ++l) {
        // Clear accumulators
        zero_kernel<<<2048, 256, 0, stream>>>((float4*)neigh_u, n_u / 4);
        zero_kernel<<<2048, 256, 0, stream>>>((float4*)neigh_i, n_i / 4);

        // neigh_u = adj_u2i @ h_i ; neigh_i = adj_i2u @ h_u
        spmm_scatter<<<4096, 256, 0, stream>>>(w_u2i, u_idx, i_idx, hi, neigh_u, N_EDGES);
        spmm_scatter<<<4096, 256, 0, stream>>>(w_i2u, i_idx, u_idx, hu, neigh_i, N_EDGES);

        float* ou = (l == N_LAYERS - 1) ? out_u : h_u1;
        float* oi = (l == N_LAYERS - 1) ? out_i : h_i1;

        dense_norm<<<N_USERS / 16, 128, 0, stream>>>(hu, neigh_u, Wu + (long)l * DIM * K2, ou, N_USERS);
        dense_norm<<<N_ITEMS / 16, 128, 0, stream>>>(hi, neigh_i, Wi + (long)l * DIM * K2, oi, N_ITEMS);

        hu = ou;
        hi = oi;
    }
}